// Net_base_11390253269707
// MI455X (gfx1250) — compile-verified
//
#include <hip/hip_runtime.h>

typedef float v2f __attribute__((ext_vector_type(2)));
typedef float v8f __attribute__((ext_vector_type(8)));

#define FDIM 32

// ---------------- degree / normalization ----------------

__global__ void k_fill_ones(float* __restrict__ p, int n) {
  int i = blockIdx.x * blockDim.x + threadIdx.x;
  if (i < n) p[i] = 1.0f;  // self-loop contributes 1 to every node's degree
}

__global__ void k_deg(const int* __restrict__ dst, float* __restrict__ deg, int e) {
  int i = blockIdx.x * blockDim.x + threadIdx.x;
  if (i < e) atomicAdd(&deg[dst[i]], 1.0f);
}

__global__ void k_rsqrt_inplace(float* __restrict__ p, int n) {
  int i = blockIdx.x * blockDim.x + threadIdx.x;
  if (i < n) p[i] = rsqrtf(p[i]);  // deg >= 1 always (self loops)
}

// ---------------- layer 1 matmul: [N,2] @ [2,32] ----------------

__global__ void k_layer1(const float* __restrict__ x, const float* __restrict__ W1,
                         float* __restrict__ h, int n) {
  int t = blockIdx.x * blockDim.x + threadIdx.x;
  if (t < n * FDIM) {
    int i = t >> 5, f = t & 31;
    h[t] = fmaf(x[2 * i], W1[f], x[2 * i + 1] * W1[FDIM + f]);
  }
}

// ---------------- aggregation: self-loop term + edge scatter ----------------

__global__ void k_self_init(const float* __restrict__ h, const float* __restrict__ dinv,
                            float* __restrict__ agg, int n) {
  int t = blockIdx.x * blockDim.x + threadIdx.x;
  if (t < n * FDIM) {
    float w = dinv[t >> 5];
    agg[t] = w * w * h[t];
  }
}

__global__ void k_edge_scatter(const int* __restrict__ src, const int* __restrict__ dst,
                               const float* __restrict__ dinv, const float* __restrict__ h,
                               float* __restrict__ agg, int e) {
  int t = blockIdx.x * blockDim.x + threadIdx.x;
  if (t < e * FDIM) {
    int ei = t >> 5, f = t & 31;
    int s = src[ei], d = dst[ei];
    float w = dinv[s] * dinv[d];
    atomicAdd(&agg[d * FDIM + f], w * h[s * FDIM + f]);
  }
}

__global__ void k_bias_relu(float* __restrict__ h, const float* __restrict__ b, int n) {
  int t = blockIdx.x * blockDim.x + threadIdx.x;
  if (t < n * FDIM) h[t] = fmaxf(h[t] + b[t & 31], 0.0f);
}

// ---------------- layer 2 matmul: [N,32] @ [32,32] via fp32 WMMA ----------------
// One wave computes a 16x32 output tile: two 16x16 accumulators, each a chain of
// eight V_WMMA_F32_16X16X4_F32 over K=32.
// A (16x4 f32) layout: lanes 0-15 hold row M=lane, VGPR0/1 = K,K+1; lanes 16-31 = K+2,K+3.
// C/D (16x16 f32) layout: VGPR r -> row r (lanes 0-15) / row r+8 (lanes 16-31), N = lane&15.
__global__ void k_gemm32_wmma(const float* __restrict__ h, const float* __restrict__ W,
                              float* __restrict__ out, int n) {
  int lane = threadIdx.x & 31;
  int tile = blockIdx.x * (blockDim.x >> 5) + (threadIdx.x >> 5);
  int row0 = tile * 16;
  if (row0 >= n) return;  // wave-uniform exit; active waves keep EXEC all-ones

  int m  = lane & 15;
  int kh = lane >> 4;  // which K-pair half this lane holds
  int rowA = row0 + m;
  if (rowA >= n) rowA = n - 1;  // clamp loads (tail); stores are guarded below

  v8f c0 = {0.f, 0.f, 0.f, 0.f, 0.f, 0.f, 0.f, 0.f};
  v8f c1 = {0.f, 0.f, 0.f, 0.f, 0.f, 0.f, 0.f, 0.f};

#pragma unroll
  for (int kk = 0; kk < 8; ++kk) {
    int k0 = kk * 4 + kh * 2;
    v2f a, b0, b1;
    a.x  = h[rowA * FDIM + k0];
    a.y  = h[rowA * FDIM + k0 + 1];
    b0.x = W[(k0)     * FDIM + m];        // output cols 0..15
    b0.y = W[(k0 + 1) * FDIM + m];
    b1.x = W[(k0)     * FDIM + 16 + m];   // output cols 16..31
    b1.y = W[(k0 + 1) * FDIM + 16 + m];
    c0 = __builtin_amdgcn_wmma_f32_16x16x4_f32(false, a, false, b0, (short)0, c0, false, false);
    c1 = __builtin_amdgcn_wmma_f32_16x16x4_f32(false, a, false, b1, (short)0, c1, false, false);
  }

  int mbase = row0 + (kh ? 8 : 0);
  if (row0 + 16 <= n) {
    // wave-uniform fast path: full tile in range, no per-row EXEC juggling
#pragma unroll
    for (int r = 0; r < 8; ++r) {
      out[(mbase + r) * FDIM + m]      = c0[r];
      out[(mbase + r) * FDIM + 16 + m] = c1[r];
    }
  } else {
    // tail tile (never taken when n % 16 == 0)
#pragma unroll
    for (int r = 0; r < 8; ++r) {
      int row = mbase + r;
      if (row < n) {
        out[row * FDIM + m]      = c0[r];
        out[row * FDIM + 16 + m] = c1[r];
      }
    }
  }
}

// ---------------- layer 3: [N,32] @ [32,1] + scalar aggregation ----------------

__global__ void k_dot32(const float* __restrict__ h, const float* __restrict__ w3,
                        float* __restrict__ o, int n) {
  int i = blockIdx.x * blockDim.x + threadIdx.x;
  if (i < n) {
    float s = 0.0f;
#pragma unroll
    for (int f = 0; f < FDIM; ++f) s = fmaf(h[i * FDIM + f], w3[f], s);
    o[i] = s;
  }
}

__global__ void k_self_init_s(const float* __restrict__ t3, const float* __restrict__ dinv,
                              const float* __restrict__ b3, float* __restrict__ out, int n) {
  int i = blockIdx.x * blockDim.x + threadIdx.x;
  if (i < n) {
    float w = dinv[i];
    out[i] = w * w * t3[i] + b3[0];
  }
}

__global__ void k_edge_scatter_s(const int* __restrict__ src, const int* __restrict__ dst,
                                 const float* __restrict__ dinv, const float* __restrict__ t3,
                                 float* __restrict__ out, int e) {
  int i = blockIdx.x * blockDim.x + threadIdx.x;
  if (i < e) {
    int s = src[i], d = dst[i];
    atomicAdd(&out[d], dinv[s] * dinv[d] * t3[s]);
  }
}

// ---------------- driver ----------------

extern "C" void kernel_launch(void* const* d_in, const int* in_sizes, int n_in,
                              void* d_out, int out_size, void* d_ws, size_t ws_size,
                              hipStream_t stream) {
  const float* x  = (const float*)d_in[0];
  const int*   ei = (const int*)d_in[1];
  const float* W1 = (const float*)d_in[2];
  const float* b1 = (const float*)d_in[3];
  const float* W2 = (const float*)d_in[4];
  const float* b2 = (const float*)d_in[5];
  const float* W3 = (const float*)d_in[6];
  const float* b3 = (const float*)d_in[7];
  float* out = (float*)d_out;

  const int N = in_sizes[0] / 2;  // x is [N,2]
  const int E = in_sizes[1] / 2;  // edge_index is [2,E]
  const int* src = ei;
  const int* dst = ei + E;

  float* dinv = (float*)d_ws;               // N   (degree, then dinv in place)
  float* hA   = dinv + N;                   // N*32
  float* hB   = hA + (size_t)N * FDIM;      // N*32
  float* t3   = hB + (size_t)N * FDIM;      // N

  const int B = 256;
  auto g = [](long long t, int b) { return (int)((t + b - 1) / b); };
  const long long nf = (long long)N * FDIM;
  const long long ef = (long long)E * FDIM;

  // symmetric GCN normalization
  k_fill_ones<<<g(N, B), B, 0, stream>>>(dinv, N);
  k_deg<<<g(E, B), B, 0, stream>>>(dst, dinv, E);
  k_rsqrt_inplace<<<g(N, B), B, 0, stream>>>(dinv, N);

  // layer 1: x@W1 -> hA; aggregate -> hB; +b1, relu
  k_layer1<<<g(nf, B), B, 0, stream>>>(x, W1, hA, N);
  k_self_init<<<g(nf, B), B, 0, stream>>>(hA, dinv, hB, N);
  k_edge_scatter<<<g(ef, B), B, 0, stream>>>(src, dst, dinv, hA, hB, E);
  k_bias_relu<<<g(nf, B), B, 0, stream>>>(hB, b1, N);

  // layer 2: hB@W2 (fp32 WMMA) -> hA; aggregate -> hB; +b2, relu
  int tiles = (N + 15) / 16;
  k_gemm32_wmma<<<g(tiles, 8), 256, 0, stream>>>(hB, W2, hA, N);
  k_self_init<<<g(nf, B), B, 0, stream>>>(hA, dinv, hB, N);
  k_edge_scatter<<<g(ef, B), B, 0, stream>>>(src, dst, dinv, hA, hB, E);
  k_bias_relu<<<g(nf, B), B, 0, stream>>>(hB, b2, N);

  // layer 3: hB@W3 -> t3; aggregate (+b3) -> out
  k_dot32<<<g(N, B), B, 0, stream>>>(hB, W3, t3, N);
  k_self_init_s<<<g(N, B), B, 0, stream>>>(t3, dinv, b3, out, N);
  k_edge_scatter_s<<<g(E, B), B, 0, stream>>>(src, dst, dinv, t3, out, E);
}